// Model_90383291777459
// MI455X (gfx1250) — compile-verified
//
#include <hip/hip_runtime.h>
#include <math.h>

// ---------------------------------------------------------------------------
// 10-layer scalar (hidden=1) tanh RNN over W=1M timesteps.
// Strategy:
//   * chunk the sequence (contraction => warmup makes chunks independent)
//   * systolic pipeline: lane l = layer l, shift-up-1 via DPP row_shr:1
//   * async global<->LDS tile staging for the x stream and output stream
// ---------------------------------------------------------------------------

#define W_TOTAL 1048576
#define NLAYERS 10
#define NCHUNK  512
#define CLEN    (W_TOTAL / NCHUNK)   // 2048
#define WARM    16384                // warmup steps; err <= max|w_hh|^WARM

#if defined(__HIP_DEVICE_COMPILE__)
  #if __has_builtin(__builtin_amdgcn_global_load_async_to_lds_b32) && \
      __has_builtin(__builtin_amdgcn_global_store_async_from_lds_b32) && \
      __has_builtin(__builtin_amdgcn_s_wait_asynccnt)
    #define HAVE_ASYNC 1
  #else
    #define HAVE_ASYNC 0
  #endif
  #if __has_builtin(__builtin_amdgcn_update_dpp)
    #define HAVE_DPP 1
  #else
    #define HAVE_DPP 0
  #endif
  #if __has_builtin(__builtin_amdgcn_tanhf)
    #define TANHF(v) __builtin_amdgcn_tanhf(v)
  #elif __has_builtin(__builtin_amdgcn_tanh_f32)
    #define TANHF(v) __builtin_amdgcn_tanh_f32(v)
  #else
    #define TANHF(v) tanhf(v)
  #endif
#else
  #define HAVE_ASYNC 0
  #define HAVE_DPP 0
  #define TANHF(v) tanhf(v)
#endif

#if defined(__HIP_DEVICE_COMPILE__)
typedef __attribute__((address_space(1))) int* gas_i32p;
typedef __attribute__((address_space(3))) int* las_i32p;
#endif

// lane l gets lane (l-1)'s value; lane 0 gets 0 (unused, it selects x).
__device__ __forceinline__ float shift_up1(float v) {
#if HAVE_DPP
  int r = __builtin_amdgcn_update_dpp(0, __builtin_bit_cast(int, v),
                                      0x111 /* row_shr:1 */, 0xf, 0xf, true);
  return __builtin_bit_cast(float, r);
#else
  return __shfl_up(v, 1, 32);
#endif
}

__device__ __forceinline__ void async_load_f32(const float* gsrc, float* lds_dst) {
#if HAVE_ASYNC
  __builtin_amdgcn_global_load_async_to_lds_b32(
      (gas_i32p)(gsrc), (las_i32p)(lds_dst), 0, 0);
#else
  *lds_dst = *gsrc;
#endif
}

__device__ __forceinline__ void async_store_f32(float* gdst, const float* lds_src) {
#if HAVE_ASYNC
  __builtin_amdgcn_global_store_async_from_lds_b32(
      (gas_i32p)(gdst), (las_i32p)(lds_src), 0, 0);
#else
  *gdst = *lds_src;
#endif
}

__device__ __forceinline__ void async_wait0() {
#if HAVE_ASYNC
  __builtin_amdgcn_s_wait_asynccnt(0);
#endif
}

// One 32-iteration tile of the systolic pipeline.
// FIRST: guard so lane l does not update h before its pipeline slot (i >= l).
template <bool FIRST>
__device__ __forceinline__ void run_tile(float& h, int lane, float wi, float wh,
                                         float bs, const float* xb, float* ob) {
#pragma unroll
  for (int j = 0; j < 32; ++j) {
    float xc  = xb[j];                       // ds_load (uniform -> broadcast)
    float hp  = shift_up1(h);                // v_mov_b32 dpp row_shr:1
    float inp = (lane == 0) ? xc : hp;
    float nh  = TANHF(__builtin_fmaf(wi, inp, __builtin_fmaf(wh, h, bs)));
    h = (!FIRST || j >= lane) ? nh : 0.0f;
    if (lane == 9) ob[j] = h;                // ds_store, EXEC-masked to lane 9
  }
}

__global__ __launch_bounds__(32) void rnn10_pipeline(
    const float* __restrict__ x,
    const float* __restrict__ w_ih, const float* __restrict__ w_hh,
    const float* __restrict__ b_ih, const float* __restrict__ b_hh,
    float* __restrict__ out) {
  __shared__ float xbuf[2][32];   // double-buffered input tiles
  __shared__ float obuf[2][32];   // double-buffered output tiles

  const int lane       = threadIdx.x;           // one wave32 per block
  const int chunk      = blockIdx.x;
  const int chunkStart = chunk * CLEN;
  const int warm       = (chunkStart < WARM) ? chunkStart : WARM;  // truncate
  const int t0         = chunkStart - warm;     // >= 0; warm==chunkStart => exact
  const int niters     = warm + CLEN + (NLAYERS - 1);
  const int ntiles     = (niters + 31) / 32;

  const int   l  = (lane < NLAYERS) ? lane : (NLAYERS - 1);
  const float wi = w_ih[l];
  const float wh = w_hh[l];
  const float bs = b_ih[l] + b_hh[l];

  float h = 0.0f;

  // prefetch tile 0 -> xbuf[0]
  {
    int gi = t0 + lane;
    gi = (gi < W_TOTAL) ? gi : (W_TOTAL - 1);
    async_load_f32(x + gi, &xbuf[0][lane]);
  }

  for (int T = 0; T < ntiles; ++T) {
    const int p = T & 1;

    async_wait0();        // xbuf[p] ready; obuf[p]'s old flush drained
    __syncthreads();      // single-wave barrier acts as compiler fence

    // prefetch tile T+1 -> xbuf[p^1]; overlaps the compute below
    {
      int gi = t0 + (T + 1) * 32 + lane;
      gi = (gi < W_TOTAL) ? gi : (W_TOTAL - 1);
      async_load_f32(x + gi, &xbuf[p ^ 1][lane]);
    }

    if (T == 0)
      run_tile<true>(h, lane, wi, wh, bs, &xbuf[p][0], &obuf[p][0]);
    else
      run_tile<false>(h, lane, wi, wh, bs, &xbuf[p][0], &obuf[p][0]);

    __syncthreads();
#if HAVE_ASYNC
    asm volatile("s_wait_dscnt 0" ::: "memory");  // lane-9 ds_stores visible
#endif
    // flush this tile's kept outputs: iteration i=T*32+lane holds timestep
    // t = t0 + i - (NLAYERS-1); keep only [chunkStart, chunkStart+CLEN)
    {
      int t = t0 + T * 32 + lane - (NLAYERS - 1);
      if (t >= chunkStart && t < chunkStart + CLEN) {
        async_store_f32(out + t, &obuf[p][lane]);
      }
    }
  }
}

extern "C" void kernel_launch(void* const* d_in, const int* in_sizes, int n_in,
                              void* d_out, int out_size, void* d_ws, size_t ws_size,
                              hipStream_t stream) {
  (void)in_sizes; (void)n_in; (void)out_size; (void)d_ws; (void)ws_size;
  const float* x    = (const float*)d_in[0];
  const float* w_ih = (const float*)d_in[1];
  const float* w_hh = (const float*)d_in[2];
  const float* b_ih = (const float*)d_in[3];
  const float* b_hh = (const float*)d_in[4];
  float* out = (float*)d_out;
  rnn10_pipeline<<<NCHUNK, 32, 0, stream>>>(x, w_ih, w_hh, b_ih, b_hh, out);
}